// LinearAttention_70042326663509
// MI455X (gfx1250) — compile-verified
//
#include <hip/hip_runtime.h>
#include <hip/hip_bf16.h>

// ---------------------------------------------------------------------------
// Linear attention (b=8, c=512, n=64*64=4096, heads=8, head_dim=64), fp32.
// All four GEMMs run on V_WMMA_F32_16X16X4_F32 (native fp32 WMMA, exact math).
// v2: b128 global staging + LDS double buffering (1 barrier / K-tile).
// ---------------------------------------------------------------------------

typedef __attribute__((ext_vector_type(2))) float v2f;
typedef __attribute__((ext_vector_type(4))) float v4f;
typedef __attribute__((ext_vector_type(8))) float v8f;

#define WAVE 32

// Generic LDS-tiled fp32 WMMA GEMM:  C[z] = scaleC * A[z](MxK) * B[z](KxN) (+bias[m])
// AT: A stored source-side as (K x M) row-major (transpose on LDS store)
// BT: B stored source-side as (N x K) row-major (transpose on LDS store)
// per-z pointer offset = (z/zdiv)*o_outer + (z%zdiv)*o_inner   (elements)
template<int MT, int NT, int KT, int WM, int WN, bool AT, bool BT, bool BIAS>
__global__ __launch_bounds__(WM * WN * WAVE)
void gemm_wmma_f32(const float* __restrict__ Ag, const float* __restrict__ Bg,
                   float* __restrict__ Cg, const float* __restrict__ bias,
                   int M, int N, int K, int lda, int ldb, int ldc,
                   int zdiv, long azo, long azi, long bzo, long bzi,
                   long czo, long czi, float scaleC)
{
    constexpr int THREADS = WM * WN * WAVE;
    constexpr int WMT = MT / WM;          // wave tile rows
    constexpr int WNT = NT / WN;          // wave tile cols
    constexpr int AM  = WMT / 16;         // 16x16 accum blocks in M per wave
    constexpr int AN  = WNT / 16;         // 16x16 accum blocks in N per wave
    constexpr int PAD = 4;                // 16B-aligned rows, conflict-free cols
    constexpr int ACH = MT * KT / 4;      // float4 chunks per A tile
    constexpr int BCH = KT * NT / 4;      // float4 chunks per B tile
    constexpr int AREG = ACH / THREADS;
    constexpr int BREG = BCH / THREADS;
    static_assert(ACH % THREADS == 0 && BCH % THREADS == 0, "tile/thread mismatch");

    __shared__ float As[2][MT][KT + PAD];
    __shared__ float Bs[2][KT][NT];

    const int tid  = threadIdx.x;
    const int wave = tid / WAVE;
    const int lane = tid % WAVE;
    const int half = lane >> 4;           // 0: lanes 0-15, 1: lanes 16-31
    const int l16  = lane & 15;
    const int wm   = (wave % WM) * WMT;
    const int wn   = (wave / WM) * WNT;

    const int z = blockIdx.z;
    const long zo = (long)(z / zdiv), zi = (long)(z % zdiv);
    const float* A = Ag + zo * azo + zi * azi;
    const float* B = Bg + zo * bzo + zi * bzi;
    float*       C = Cg + zo * czo + zi * czi;

    const int m0 = blockIdx.y * MT;
    const int n0 = blockIdx.x * NT;

    v4f areg[AREG], breg[BREG];

    // ---- register-staged global loads (b128, coalesced along source dim) ----
    auto loadA = [&](int k0) {
#pragma unroll
        for (int r = 0; r < AREG; ++r) {
            const int L = (tid + r * THREADS) * 4;
            if (AT) {
                const int mm = L % MT, kk = L / MT;
                areg[r] = *(const v4f*)&A[(long)(k0 + kk) * lda + (m0 + mm)];
            } else {
                const int kk = L % KT, mm = L / KT;
                areg[r] = *(const v4f*)&A[(long)(m0 + mm) * lda + (k0 + kk)];
            }
        }
    };
    auto loadB = [&](int k0) {
#pragma unroll
        for (int r = 0; r < BREG; ++r) {
            const int L = (tid + r * THREADS) * 4;
            if (BT) {
                const int kk = L % KT, nn = L / KT;
                breg[r] = *(const v4f*)&B[(long)(n0 + nn) * ldb + (k0 + kk)];
            } else {
                const int nn = L % NT, kk = L / NT;
                breg[r] = *(const v4f*)&B[(long)(k0 + kk) * ldb + (n0 + nn)];
            }
        }
    };
    auto storeA = [&](int buf) {
#pragma unroll
        for (int r = 0; r < AREG; ++r) {
            const int L = (tid + r * THREADS) * 4;
            if (AT) {
                const int mm = L % MT, kk = L / MT;
#pragma unroll
                for (int j = 0; j < 4; ++j) As[buf][mm + j][kk] = areg[r][j];
            } else {
                const int kk = L % KT, mm = L / KT;
                *(v4f*)&As[buf][mm][kk] = areg[r];
            }
        }
    };
    auto storeB = [&](int buf) {
#pragma unroll
        for (int r = 0; r < BREG; ++r) {
            const int L = (tid + r * THREADS) * 4;
            if (BT) {
                const int kk = L % KT, nn = L / KT;
#pragma unroll
                for (int j = 0; j < 4; ++j) Bs[buf][kk + j][nn] = breg[r][j];
            } else {
                const int nn = L % NT, kk = L / NT;
                *(v4f*)&Bs[buf][kk][nn] = breg[r];
            }
        }
    };

    v8f acc[AM][AN];
#pragma unroll
    for (int i = 0; i < AM; ++i)
#pragma unroll
        for (int j = 0; j < AN; ++j)
            acc[i][j] = (v8f){0.f, 0.f, 0.f, 0.f, 0.f, 0.f, 0.f, 0.f};

    const int nk = K / KT;
    loadA(0); loadB(0);
    storeA(0); storeB(0);
    int buf = 0;

    for (int t = 0; t < nk; ++t) {
        __syncthreads();                          // tile t visible in LDS buffer buf
        if (t + 1 < nk) {                         // prefetch t+1 while WMMAs run
            loadA((t + 1) * KT);
            loadB((t + 1) * KT);
        }

        // ---- WMMA over the K tile, 4 at a time ----
#pragma unroll
        for (int k4 = 0; k4 < KT; k4 += 4) {
            const int kb = k4 + 2 * half;         // ISA 32-bit A layout: lanes
            v2f afr[AM], bfr[AN];                 // 0-15 -> K{0,1}, 16-31 -> K{2,3}
#pragma unroll
            for (int i = 0; i < AM; ++i)
                afr[i] = (v2f){As[buf][wm + i * 16 + l16][kb],
                               As[buf][wm + i * 16 + l16][kb + 1]};
#pragma unroll
            for (int j = 0; j < AN; ++j)
                bfr[j] = (v2f){Bs[buf][kb][wn + j * 16 + l16],
                               Bs[buf][kb + 1][wn + j * 16 + l16]};
#pragma unroll
            for (int i = 0; i < AM; ++i)
#pragma unroll
                for (int j = 0; j < AN; ++j)
                    acc[i][j] = __builtin_amdgcn_wmma_f32_16x16x4_f32(
                        false, afr[i], false, bfr[j], (short)0, acc[i][j],
                        false, false);
        }

        if (t + 1 < nk) {                         // write t+1 into the idle buffer
            storeA(buf ^ 1);
            storeB(buf ^ 1);
        }
        buf ^= 1;
    }

    // ---- epilogue: C/D layout — VGPR r holds M = r + 8*half, N = lane%16 ----
#pragma unroll
    for (int i = 0; i < AM; ++i)
#pragma unroll
        for (int j = 0; j < AN; ++j) {
            const int col = n0 + wn + j * 16 + l16;
#pragma unroll
            for (int r = 0; r < 8; ++r) {
                const int row = m0 + wm + i * 16 + half * 8 + r;
                float v = acc[i][j][r] * scaleC;
                if (BIAS) v += bias[row];
                C[(long)row * ldc + col] = v;
            }
        }
}

// softmax over head_dim (64) for each (b,h,n) column of q; folds scale=1/8.
__global__ __launch_bounds__(256) void softmax_q_kernel(float* qkv, float scale)
{
    const int z = blockIdx.y;                    // b*8 + h
    const int b = z >> 3, h = z & 7;
    const int n = blockIdx.x * 256 + threadIdx.x;
    float* base = qkv + (long)b * 1536 * 4096 + (long)h * 64 * 4096 + n;

    float vals[64];
    float m = -3.4e38f;
#pragma unroll
    for (int d = 0; d < 64; ++d) {
        vals[d] = base[(long)d * 4096];
        m = fmaxf(m, vals[d]);
    }
    float s = 0.f;
#pragma unroll
    for (int d = 0; d < 64; ++d) {
        vals[d] = __expf(vals[d] - m);
        s += vals[d];
    }
    const float r = scale / s;
#pragma unroll
    for (int d = 0; d < 64; ++d)
        base[(long)d * 4096] = vals[d] * r;
}

// softmax over tokens (4096) for each row of k. One block per row.
__global__ __launch_bounds__(256) void softmax_k_kernel(float* qkv)
{
    const int r = blockIdx.x;                    // 0..4095 = b*512 + c
    const int b = r >> 9, c = r & 511;
    float* row = qkv + (long)b * 1536 * 4096 + (long)(512 + c) * 4096;

    __shared__ float red[256];
    const int tid = threadIdx.x;

    float v[16];
    float m = -3.4e38f;
#pragma unroll
    for (int i = 0; i < 16; ++i) {
        v[i] = row[tid + i * 256];
        m = fmaxf(m, v[i]);
    }
    red[tid] = m;
    __syncthreads();
    for (int s = 128; s > 0; s >>= 1) {
        if (tid < s) red[tid] = fmaxf(red[tid], red[tid + s]);
        __syncthreads();
    }
    m = red[0];
    __syncthreads();

    float sum = 0.f;
#pragma unroll
    for (int i = 0; i < 16; ++i) {
        v[i] = __expf(v[i] - m);
        sum += v[i];
    }
    red[tid] = sum;
    __syncthreads();
    for (int s = 128; s > 0; s >>= 1) {
        if (tid < s) red[tid] += red[tid + s];
        __syncthreads();
    }
    const float inv = 1.f / red[0];
#pragma unroll
    for (int i = 0; i < 16; ++i)
        row[tid + i * 256] = v[i] * inv;
}

// Channel LayerNorm over c=512 for each (b,n) column; in-place on y, scale g.
__global__ __launch_bounds__(256) void layernorm_kernel(float* y, const float* __restrict__ g)
{
    const int col = blockIdx.x * 256 + threadIdx.x;   // 0..32767
    const int b = col >> 12, n = col & 4095;
    float* base = y + (long)b * 512 * 4096 + n;

    float s = 0.f, s2 = 0.f;
    for (int c = 0; c < 512; ++c) {
        const float t = base[(long)c * 4096];
        s += t;
        s2 += t * t;
    }
    const float mean = s * (1.f / 512.f);
    const float var  = s2 * (1.f / 512.f) - mean * mean;
    const float rs   = rsqrtf(var + 1e-5f);
    for (int c = 0; c < 512; ++c)
        base[(long)c * 4096] = (base[(long)c * 4096] - mean) * rs * g[c];
}

extern "C" void kernel_launch(void* const* d_in, const int* in_sizes, int n_in,
                              void* d_out, int out_size, void* d_ws, size_t ws_size,
                              hipStream_t stream) {
    const float* x     = (const float*)d_in[0];   // (8, 512, 64, 64)
    const float* w_qkv = (const float*)d_in[1];   // (1536, 512)
    const float* w_out = (const float*)d_in[2];   // (512, 512)
    const float* b_out = (const float*)d_in[3];   // (512,)
    const float* g     = (const float*)d_in[4];   // (512,)
    float* y = (float*)d_out;                     // (8, 512, 64, 64)

    // workspace layout (floats)
    float* qkv  = (float*)d_ws;                   // 8*1536*4096 = 50331648
    float* outb = qkv + (long)8 * 1536 * 4096;    // 8*512*4096  = 16777216
    float* ctx  = outb + (long)8 * 512 * 4096;    // 8*8*64*64   = 262144

    const long S = (long)512 * 4096;              // one batch slab of 512 rows

    // 1) qkv = w_qkv (1536x512) @ x[b] (512x4096)
    gemm_wmma_f32<64, 128, 16, 2, 4, false, false, false>
        <<<dim3(32, 24, 8), 256, 0, stream>>>(
            w_qkv, x, qkv, nullptr,
            1536, 4096, 512, 512, 4096, 4096,
            1, 0, 0, S, 0, 3 * S, 0, 1.0f);

    // 2) softmax over head_dim on q (rows 0..511 of each batch), fold scale=1/8
    softmax_q_kernel<<<dim3(16, 64), 256, 0, stream>>>(qkv, 0.125f);

    // 3) softmax over tokens on k (rows 512..1023 of each batch)
    softmax_k_kernel<<<dim3(4096), 256, 0, stream>>>(qkv);

    // 4) context[b,h] (64x64) = k (64x4096) @ v^T (4096x64), scaled by 1/4096
    gemm_wmma_f32<64, 64, 32, 4, 2, false, true, false>
        <<<dim3(1, 1, 64), 256, 0, stream>>>(
            qkv + S, qkv + 2 * S, ctx, nullptr,
            64, 64, 4096, 4096, 4096, 64,
            8, 3 * S, (long)64 * 4096, 3 * S, (long)64 * 4096,
            (long)8 * 64 * 64, (long)64 * 64, 1.0f / 4096.f);

    // 5) out[b,h] (64x4096) = context^T (64x64) @ q (64x4096)
    gemm_wmma_f32<64, 128, 16, 2, 4, true, false, false>
        <<<dim3(32, 1, 64), 256, 0, stream>>>(
            ctx, qkv, outb, nullptr,
            64, 4096, 64, 64, 4096, 4096,
            8, (long)8 * 64 * 64, (long)64 * 64, 3 * S, (long)64 * 4096,
            S, (long)64 * 4096, 1.0f);

    // 6) y[b] = w_out (512x512) @ out[b] (512x4096) + b_out
    gemm_wmma_f32<64, 128, 16, 2, 4, false, false, true>
        <<<dim3(32, 8, 8), 256, 0, stream>>>(
            w_out, outb, y, b_out,
            512, 4096, 512, 512, 4096, 4096,
            1, 0, 0, S, 0, S, 0, 1.0f);

    // 7) channel LayerNorm in-place on y
    layernorm_kernel<<<dim3(128), 256, 0, stream>>>(y, g);
}